// OctreeInterp_17188459119028
// MI455X (gfx1250) — compile-verified
//
#include <hip/hip_runtime.h>
#include <hip/hip_bf16.h>

// Octree trilinear interpolation for MI455X (gfx1250, wave32).
//
// Inputs (setup_inputs order):
//   d_in[0] data      : float32 [H, 32]
//   d_in[1] pts       : float32 [N, 4]  (x,y,z,batch)
//   d_in[2] node_keys : int32   [H]     (sorted)
//   d_in[3] depth     : int32   [1]
// Output: float32 [N, 32]
//
// One thread per point. LUT narrows each search to a ~31-entry bucket; the 8
// corner searches run branchless in lockstep so all 8 probe loads issue per
// iteration (single wait), maximizing MLP on the L2-latency-bound path.
// Corners whose key is absent (≈97% for random keys) contribute w=0 and skip
// the 128B feature gather entirely — identical math, ~30x less gather traffic.

#define LUT_BITS 16
#define LUT_SIZE (1 << LUT_BITS)          // 65536 buckets, 65537 boundary entries

__device__ __forceinline__ int spread3(int v, int cmask) {
  // Spread low bits of v (masked to depth bits, depth <= 10) to every 3rd bit.
  unsigned x = (unsigned)(v & cmask);
  x = (x | (x << 16)) & 0x030000FFu;
  x = (x | (x <<  8)) & 0x0300F00Fu;
  x = (x | (x <<  4)) & 0x030C30C3u;
  x = (x | (x <<  2)) & 0x09249249u;
  return (int)x;
}

// lut[e] = lower_bound(node_keys, e << shift), e in [0, LUT_SIZE]
__global__ __launch_bounds__(256) void octree_build_lut_kernel(
    const int* __restrict__ node_keys, int H,
    const int* __restrict__ depth_ptr, int* __restrict__ lut) {
  const int e = blockIdx.x * blockDim.x + threadIdx.x;
  if (e > LUT_SIZE) return;
  const int depth = *depth_ptr;
  const int kbits = 3 * depth + 2;                     // batch B=4 -> 2 high bits
  const int shift = (kbits > LUT_BITS) ? (kbits - LUT_BITS) : 0;
  const int target = e << shift;                       // fits int32 for depth <= 9
  int lo = 0, hi = H;
  while (lo < hi) {
    int mid = (lo + hi) >> 1;
    if (node_keys[mid] < target) lo = mid + 1; else hi = mid;
  }
  lut[e] = lo;
}

__global__ __launch_bounds__(256) void octree_interp_kernel(
    const float* __restrict__ data, const float* __restrict__ pts,
    const int* __restrict__ node_keys, const int* __restrict__ depth_ptr,
    const int* __restrict__ lut,      // may be nullptr -> full-range searches
    float* __restrict__ out, int N, int H) {
  const int n = blockIdx.x * blockDim.x + threadIdx.x;
  if (n >= N) return;

  const int depth = *depth_ptr;
  const int cmask = (1 << depth) - 1;
  const int kbits = 3 * depth + 2;
  const int shift = (kbits > LUT_BITS) ? (kbits - LUT_BITS) : 0;

  const float4 pt = reinterpret_cast<const float4*>(pts)[n];
  const float scale = (float)(1 << (depth - 1));

  const float xf = (pt.x + 1.0f) * scale - 0.5f;
  const float yf = (pt.y + 1.0f) * scale - 0.5f;
  const float zf = (pt.z + 1.0f) * scale - 0.5f;
  const float fxi = floorf(xf), fyi = floorf(yf), fzi = floorf(zf);
  const int xi = (int)fxi, yi = (int)fyi, zi = (int)fzi;
  const float fx = xf - fxi, fy = yf - fyi, fz = zf - fzi;
  const int bterm = ((int)pt.w) << (3 * depth);

  // 6 distinct Morton spreads cover all 8 corner keys.
  const int sx0 = spread3(xi,     cmask) << 2, sx1 = spread3(xi + 1, cmask) << 2;
  const int sy0 = spread3(yi,     cmask) << 1, sy1 = spread3(yi + 1, cmask) << 1;
  const int sz0 = spread3(zi,     cmask),      sz1 = spread3(zi + 1, cmask);

  int key[8], lo[8], len[8];
  int maxlen = 0;
#pragma unroll
  for (int p = 0; p < 8; ++p) {
    key[p] = bterm | ((p & 4) ? sx1 : sx0) | ((p & 2) ? sy1 : sy0)
                   | ((p & 1) ? sz1 : sz0);
    int l = 0, h = H;
    if (lut) {
      unsigned e = ((unsigned)key[p]) >> shift;
      if (e < (unsigned)LUT_SIZE) { l = lut[e]; h = lut[e + 1]; }
    }
    lo[p]  = l;
    len[p] = h - l;
    maxlen = (len[p] > maxlen) ? len[p] : maxlen;
  }

  // Branchless lockstep lower_bound: every iteration issues 8 independent
  // probe loads (no per-probe branches), then pure select updates.
  for (; maxlen > 0; maxlen >>= 1) {
#pragma unroll
    for (int p = 0; p < 8; ++p) {
      const int half = len[p] >> 1;
      const int mid  = lo[p] + half;
      const int safe = (mid < H - 1) ? mid : (H - 1);
      const int v    = node_keys[safe];
      const bool pred = (len[p] > 0) && (v < key[p]);
      lo[p]  = pred ? (mid + 1) : lo[p];
      len[p] = pred ? (len[p] - half - 1) : half;
    }
  }

  float4 acc[8];
#pragma unroll
  for (int c = 0; c < 8; ++c) acc[c] = make_float4(0.f, 0.f, 0.f, 0.f);

  float norm = 0.0f;
#pragma unroll
  for (int p = 0; p < 8; ++p) {
    const int pos = (lo[p] < H - 1) ? lo[p] : (H - 1);
    const bool valid = (node_keys[pos] == key[p]);
    float wp = ((p & 4) ? fx : 1.0f - fx) *
               ((p & 2) ? fy : 1.0f - fy) *
               ((p & 1) ? fz : 1.0f - fz);
    wp = valid ? wp : 0.0f;
    norm += wp;
    if (wp != 0.0f) {
      const float4* __restrict__ row =
          reinterpret_cast<const float4*>(data + (size_t)pos * 32u);
      __builtin_prefetch(row, 0, 3);   // global_prefetch_b8
#pragma unroll
      for (int c = 0; c < 8; ++c) {
        const float4 v = row[c];
        acc[c].x += wp * v.x;
        acc[c].y += wp * v.y;
        acc[c].z += wp * v.z;
        acc[c].w += wp * v.w;
      }
    }
  }

  const float inv = 1.0f / (norm + 1e-12f);
  float4* __restrict__ o = reinterpret_cast<float4*>(out + (size_t)n * 32u);
#pragma unroll
  for (int c = 0; c < 8; ++c) {
    o[c] = make_float4(acc[c].x * inv, acc[c].y * inv,
                       acc[c].z * inv, acc[c].w * inv);
  }
}

extern "C" void kernel_launch(void* const* d_in, const int* in_sizes, int n_in,
                              void* d_out, int out_size, void* d_ws, size_t ws_size,
                              hipStream_t stream) {
  const float* data      = (const float*)d_in[0];
  const float* pts       = (const float*)d_in[1];
  const int*   node_keys = (const int*)d_in[2];
  const int*   depth_ptr = (const int*)d_in[3];
  float*       out       = (float*)d_out;

  const int H = in_sizes[2];
  const int N = in_sizes[1] / 4;

  int* lut = nullptr;
  if (ws_size >= (size_t)(LUT_SIZE + 1) * sizeof(int)) {
    lut = (int*)d_ws;
    const int lut_threads = LUT_SIZE + 1;
    octree_build_lut_kernel<<<(lut_threads + 255) / 256, 256, 0, stream>>>(
        node_keys, H, depth_ptr, lut);
  }

  octree_interp_kernel<<<(N + 255) / 256, 256, 0, stream>>>(
      data, pts, node_keys, depth_ptr, lut, out, N, H);
}